// Net_SGC_11227044511902
// MI455X (gfx1250) — compile-verified
//
#include <hip/hip_runtime.h>
#include <hip/hip_bf16.h>
#include <math.h>

typedef __attribute__((ext_vector_type(2))) float v2f;
typedef __attribute__((ext_vector_type(8))) float v8f;

#define HDIM 64
#define GNUM 10
#define CNUM 10

// ---------------- degree / normalization ----------------
__global__ void deg_init_kernel(float* __restrict__ deg, int n) {
    int i = blockIdx.x * blockDim.x + threadIdx.x;
    if (i < n) deg[i] = 1.0f;  // self-loop weight
}

__global__ void deg_acc_kernel(const int* __restrict__ col,
                               const float* __restrict__ ew,
                               float* __restrict__ deg, int e) {
    int i = blockIdx.x * blockDim.x + threadIdx.x;
    if (i < e) atomicAdd(&deg[col[i]], ew[i]);
}

__global__ void dinv_kernel(float* __restrict__ deg_to_dinv, int n) {
    int i = blockIdx.x * blockDim.x + threadIdx.x;
    if (i < n) {
        float d = deg_to_dinv[i];
        deg_to_dinv[i] = (d > 0.0f) ? rsqrtf(d) : 0.0f;
    }
}

// ---------------- CSR build (push -> pull conversion) ----------------
__global__ void zero_int_kernel(int* __restrict__ p, int n) {
    int i = blockIdx.x * blockDim.x + threadIdx.x;
    if (i < n) p[i] = 0;
}

__global__ void cnt_kernel(const int* __restrict__ col, int* __restrict__ cnt, int e) {
    int i = blockIdx.x * blockDim.x + threadIdx.x;
    if (i < e) atomicAdd(&cnt[col[i]], 1);
}

// per-256-block exclusive scan; block totals to bsums
__global__ __launch_bounds__(256)
void scan_block_kernel(const int* __restrict__ cnt, int* __restrict__ rowptr,
                       int* __restrict__ bsums, int n) {
    __shared__ int tmp[256];
    int t = threadIdx.x;
    int idx = blockIdx.x * 256 + t;
    int v = (idx < n) ? cnt[idx] : 0;
    tmp[t] = v;
    __syncthreads();
    for (int off = 1; off < 256; off <<= 1) {
        int x = (t >= off) ? tmp[t - off] : 0;
        __syncthreads();
        tmp[t] += x;
        __syncthreads();
    }
    if (idx < n) rowptr[idx] = tmp[t] - v;   // exclusive within block
    if (t == 255) bsums[blockIdx.x] = tmp[255];
}

// single-block exclusive scan of up to 1024 block sums
__global__ __launch_bounds__(1024)
void scan_sums_kernel(int* __restrict__ bs, int nb) {
    __shared__ int tmp[1024];
    int t = threadIdx.x;
    int v = (t < nb) ? bs[t] : 0;
    tmp[t] = v;
    __syncthreads();
    for (int off = 1; off < 1024; off <<= 1) {
        int x = (t >= off) ? tmp[t - off] : 0;
        __syncthreads();
        tmp[t] += x;
        __syncthreads();
    }
    if (t < nb) bs[t] = tmp[t] - v;          // exclusive
}

__global__ void scan_add_kernel(int* __restrict__ rowptr, const int* __restrict__ bsums,
                                int n, int e) {
    int idx = blockIdx.x * blockDim.x + threadIdx.x;
    if (idx < n) rowptr[idx] += bsums[idx >> 8];
    if (idx == 0) rowptr[n] = e;
}

__global__ void scatter_kernel(const int* __restrict__ row, const int* __restrict__ col,
                               const float* __restrict__ ew, const float* __restrict__ dinv,
                               const int* __restrict__ rowptr, int* __restrict__ fill,
                               int* __restrict__ csr_src, float* __restrict__ csr_w, int e) {
    int i = blockIdx.x * blockDim.x + threadIdx.x;
    if (i < e) {
        int c = col[i], r = row[i];
        int p = rowptr[c] + atomicAdd(&fill[c], 1);
        csr_src[p] = r;
        csr_w[p] = dinv[r] * ew[i] * dinv[c];
    }
}

// ---------------- pull-based hop: one wave32 per destination node ----------------
// acc initialized with self-loop term dinv^2 * h_in[node]; no float atomics.
__global__ __launch_bounds__(256)
void hop_gather_kernel(const int* __restrict__ rowptr, const int* __restrict__ csr_src,
                       const float* __restrict__ csr_w, const float* __restrict__ dinv,
                       const float* __restrict__ hin, float* __restrict__ hout, int n) {
    int node = blockIdx.x * (blockDim.x >> 5) + (threadIdx.x >> 5);
    int lane = threadIdx.x & 31;
    if (node >= n) return;  // wave-uniform
    float di = dinv[node];
    float2 v0 = ((const float2*)(hin + (long long)node * HDIM))[lane];
    float2 acc;
    acc.x = di * di * v0.x;
    acc.y = di * di * v0.y;
    int s = rowptr[node], t = rowptr[node + 1];
    for (int base = s; base < t; base += 32) {
        int j = base + lane;
        int rr = (j < t) ? csr_src[j] : 0;       // coalesced metadata load
        float wwv = (j < t) ? csr_w[j] : 0.0f;
        int m = t - base; if (m > 32) m = 32;
        for (int k = 0; k < m; ++k) {
            int r = __shfl(rr, k);
            float w = __shfl(wwv, k);
            float2 v = ((const float2*)(hin + (long long)r * HDIM))[lane];
            acc.x += w * v.x;
            acc.y += w * v.y;
        }
    }
    ((float2*)(hout + (long long)node * HDIM))[lane] = acc;
}

// ---------------- fallback atomic-scatter hop (used only if ws too small) ----------
__global__ void hop_init_kernel(const float* __restrict__ dinv,
                                const float* __restrict__ hin,
                                float* __restrict__ hout, int n) {
    long long idx = (long long)blockIdx.x * blockDim.x + threadIdx.x;
    long long total = (long long)n * HDIM;
    if (idx < total) {
        int node = (int)(idx >> 6);
        float di = dinv[node];
        hout[idx] = di * di * hin[idx];
    }
}

__global__ __launch_bounds__(256)
void spmm_edge_kernel(const int* __restrict__ row, const int* __restrict__ col,
                      const float* __restrict__ ew, const float* __restrict__ dinv,
                      const float* __restrict__ hin, float* __restrict__ hout, int e) {
    int warp = blockIdx.x * (blockDim.x >> 5) + (threadIdx.x >> 5);
    int lane = threadIdx.x & 31;
    if (warp >= e) return;
    int r = row[warp];
    int c = col[warp];
    float w = dinv[r] * ew[warp] * dinv[c];
    float2 v = ((const float2*)(hin + (long long)r * HDIM))[lane];
    float* dst = hout + (long long)c * HDIM + lane * 2;
    atomicAdd(dst + 0, w * v.x);
    atomicAdd(dst + 1, w * v.y);
}

// ---------------- dense GEMM via V_WMMA_F32_16X16X4_F32, W staged in LDS ----------
__global__ __launch_bounds__(256)
void gemm_wmma_kernel(const float* __restrict__ h,
                      const float* __restrict__ W,
                      const float* __restrict__ bias,
                      float* __restrict__ out, int n) {
    __shared__ float sW[HDIM * HDIM];        // 16 KB
    {   // cooperative stage: 4096 floats = 1024 float4, 4 per thread
        const float4* Wv = (const float4*)W;
        float4* sWv = (float4*)sW;
        for (int i = threadIdx.x; i < (HDIM * HDIM) / 4; i += blockDim.x)
            sWv[i] = Wv[i];
    }
    __syncthreads();                          // before any wave-uniform exit

    int wave = blockIdx.x * (blockDim.x >> 5) + (threadIdx.x >> 5);
    int lane = threadIdx.x & 31;
    const int tilesN = HDIM / 16;             // 4
    int tM = wave / tilesN;
    int tN = wave % tilesN;
    if (tM * 16 >= n) return;                 // wave-uniform: EXEC all-ones for WMMA

    int l15 = lane & 15;
    int hi  = lane >> 4;

    int mA = tM * 16 + l15;
    if (mA >= n) mA = n - 1;
    const float* Arow = h + (long long)mA * HDIM + hi * 2;

    v8f acc = {};
#pragma unroll
    for (int k = 0; k < HDIM; k += 4) {
        v2f a;
        a.x = Arow[k];
        a.y = Arow[k + 1];
        v2f bm;
        bm.x = sW[(k + hi * 2 + 0) * HDIM + tN * 16 + l15];
        bm.y = sW[(k + hi * 2 + 1) * HDIM + tN * 16 + l15];
        acc = __builtin_amdgcn_wmma_f32_16x16x4_f32(
            false, a, false, bm, (short)0, acc, false, false);
    }

    float bv = bias[tN * 16 + l15];
#pragma unroll
    for (int j = 0; j < 8; ++j) {
        int m = tM * 16 + j + hi * 8;
        if (m < n)
            out[(long long)m * HDIM + tN * 16 + l15] = acc[j] + bv;
    }
}

// ---------------- pooling ----------------
__global__ void pool_init_kernel(float* __restrict__ sums, float* __restrict__ cnt) {
    int i = blockIdx.x * blockDim.x + threadIdx.x;
    if (i < GNUM * HDIM) sums[i] = 0.0f;
    if (i < GNUM) cnt[i] = 0.0f;
}

__global__ __launch_bounds__(256)
void pool_kernel(const float* __restrict__ h, const int* __restrict__ batch,
                 float* __restrict__ sums, float* __restrict__ cnt, int n) {
    __shared__ float ls[GNUM * HDIM];
    __shared__ float lc[GNUM];
    for (int i = threadIdx.x; i < GNUM * HDIM; i += blockDim.x) ls[i] = 0.0f;
    if (threadIdx.x < GNUM) lc[threadIdx.x] = 0.0f;
    __syncthreads();

    int f   = threadIdx.x & (HDIM - 1);
    int sub = threadIdx.x >> 6;
    const int chunk = 1024;
    int start = blockIdx.x * chunk;
    int end   = start + chunk; if (end > n) end = n;
    for (int node = start + sub; node < end; node += 4) {
        int g = batch[node];
        atomicAdd(&ls[g * HDIM + f], h[(long long)node * HDIM + f]);
        if (f == 0) atomicAdd(&lc[g], 1.0f);
    }
    __syncthreads();
    for (int i = threadIdx.x; i < GNUM * HDIM; i += blockDim.x)
        if (ls[i] != 0.0f) atomicAdd(&sums[i], ls[i]);
    if (threadIdx.x < GNUM && lc[threadIdx.x] != 0.0f)
        atomicAdd(&cnt[threadIdx.x], lc[threadIdx.x]);
}

// ---------------- MLP head + log_softmax ----------------
__global__ __launch_bounds__(128)
void head_kernel(const float* __restrict__ sums, const float* __restrict__ cnt,
                 const float* __restrict__ w1, const float* __restrict__ b1,
                 const float* __restrict__ w2, const float* __restrict__ b2,
                 float* __restrict__ out) {
    __shared__ float gm[GNUM * HDIM];
    __shared__ float h1[GNUM * HDIM];
    __shared__ float lg[GNUM * CNUM];
    int t = threadIdx.x;

    for (int i = t; i < GNUM * HDIM; i += blockDim.x) {
        int g = i >> 6;
        gm[i] = sums[i] / fmaxf(cnt[g], 1.0f);
    }
    __syncthreads();

    for (int i = t; i < GNUM * HDIM; i += blockDim.x) {
        int g = i >> 6, j = i & (HDIM - 1);
        float s = b1[j];
#pragma unroll 8
        for (int k = 0; k < HDIM; ++k) s += gm[g * HDIM + k] * w1[k * HDIM + j];
        h1[i] = fmaxf(s, 0.0f);
    }
    __syncthreads();

    for (int i = t; i < GNUM * CNUM; i += blockDim.x) {
        int g = i / CNUM, j = i % CNUM;
        float s = b2[j];
#pragma unroll 8
        for (int k = 0; k < HDIM; ++k) s += h1[g * HDIM + k] * w2[k * CNUM + j];
        lg[i] = s;
    }
    __syncthreads();

    if (t < GNUM) {
        float m = -1e30f;
        for (int j = 0; j < CNUM; ++j) m = fmaxf(m, lg[t * CNUM + j]);
        float s = 0.0f;
        for (int j = 0; j < CNUM; ++j) s += expf(lg[t * CNUM + j] - m);
        float lse = m + logf(s);
        for (int j = 0; j < CNUM; ++j) out[t * CNUM + j] = lg[t * CNUM + j] - lse;
    }
}

// ---------------- launcher ----------------
extern "C" void kernel_launch(void* const* d_in, const int* in_sizes, int n_in,
                              void* d_out, int out_size, void* d_ws, size_t ws_size,
                              hipStream_t stream) {
    const float* x      = (const float*)d_in[0];
    const int*   eidx   = (const int*)  d_in[1];
    const float* ew     = (const float*)d_in[2];
    const int*   batch  = (const int*)  d_in[3];
    const float* conv_w = (const float*)d_in[4];
    const float* conv_b = (const float*)d_in[5];
    const float* lin1_w = (const float*)d_in[6];
    const float* lin1_b = (const float*)d_in[7];
    const float* lin2_w = (const float*)d_in[8];
    const float* lin2_b = (const float*)d_in[9];
    float* out = (float*)d_out;

    const int N = in_sizes[0] / HDIM;
    const int E = in_sizes[1] / 2;
    const int* row = eidx;
    const int* col = eidx + E;
    const int nb = (N + 255) / 256;           // <= 1024 for N <= 262144

    // workspace layout
    float* hA   = (float*)d_ws;                           // N*64 f
    float* hB   = hA + (long long)N * HDIM;               // N*64 f
    float* dinv = hB + (long long)N * HDIM;               // N f
    float* sums = dinv + N;                               // G*64 f
    float* cnt  = sums + GNUM * HDIM;                     // G f
    int*   rowptr = (int*)(cnt + GNUM);                   // N+1 i
    int*   fill   = rowptr + (N + 1);                     // N i (counts, then fill)
    int*   bsums  = fill + N;                             // nb i
    int*   csr_src= bsums + nb;                           // E i
    float* csr_w  = (float*)(csr_src + E);                // E f

    size_t need_csr = (size_t)((char*)(csr_w + E) - (char*)d_ws);
    bool use_csr = (ws_size >= need_csr) && (nb <= 1024);

    // 1. degree + dinv (dinv buffer holds deg, then dinv in-place)
    deg_init_kernel<<<(N + 255) / 256, 256, 0, stream>>>(dinv, N);
    deg_acc_kernel<<<(E + 255) / 256, 256, 0, stream>>>(col, ew, dinv, E);
    dinv_kernel<<<(N + 255) / 256, 256, 0, stream>>>(dinv, N);

    const int nodeWaveBlocks = (N + 7) / 8;   // 8 waves/block, 1 wave/node
    const int edgeBlocks256  = (E + 255) / 256;

    if (use_csr) {
        // 2. build destination-ordered CSR
        zero_int_kernel<<<(N + 255) / 256, 256, 0, stream>>>(fill, N);
        cnt_kernel<<<edgeBlocks256, 256, 0, stream>>>(col, fill, E);
        scan_block_kernel<<<nb, 256, 0, stream>>>(fill, rowptr, bsums, N);
        scan_sums_kernel<<<1, 1024, 0, stream>>>(bsums, nb);
        scan_add_kernel<<<(N + 255) / 256, 256, 0, stream>>>(rowptr, bsums, N, E);
        zero_int_kernel<<<(N + 255) / 256, 256, 0, stream>>>(fill, N);
        scatter_kernel<<<edgeBlocks256, 256, 0, stream>>>(row, col, ew, dinv,
                                                          rowptr, fill, csr_src, csr_w, E);
        // 3. K=3 pull-based hops (x -> hA -> hB -> hA), no float atomics
        hop_gather_kernel<<<nodeWaveBlocks, 256, 0, stream>>>(rowptr, csr_src, csr_w, dinv, x,  hA, N);
        hop_gather_kernel<<<nodeWaveBlocks, 256, 0, stream>>>(rowptr, csr_src, csr_w, dinv, hA, hB, N);
        hop_gather_kernel<<<nodeWaveBlocks, 256, 0, stream>>>(rowptr, csr_src, csr_w, dinv, hB, hA, N);
    } else {
        // fallback: atomic scatter hops
        const long long totF = (long long)N * HDIM;
        const int initBlocks = (int)((totF + 255) / 256);
        const int edgeWaveBlocks = (E + 7) / 8;
        hop_init_kernel<<<initBlocks, 256, 0, stream>>>(dinv, x, hA, N);
        spmm_edge_kernel<<<edgeWaveBlocks, 256, 0, stream>>>(row, col, ew, dinv, x, hA, E);
        hop_init_kernel<<<initBlocks, 256, 0, stream>>>(dinv, hA, hB, N);
        spmm_edge_kernel<<<edgeWaveBlocks, 256, 0, stream>>>(row, col, ew, dinv, hA, hB, E);
        hop_init_kernel<<<initBlocks, 256, 0, stream>>>(dinv, hB, hA, N);
        spmm_edge_kernel<<<edgeWaveBlocks, 256, 0, stream>>>(row, col, ew, dinv, hB, hA, E);
    }

    // 4. SGConv linear: hB = hA @ conv_w + conv_b (WMMA f32 16x16x4, W in LDS)
    const int tilesM = (N + 15) / 16;
    const int waves  = tilesM * (HDIM / 16);
    gemm_wmma_kernel<<<(waves + 7) / 8, 256, 0, stream>>>(hA, conv_w, conv_b, hB, N);

    // 5. global mean pool
    pool_init_kernel<<<(GNUM * HDIM + 255) / 256, 256, 0, stream>>>(sums, cnt);
    pool_kernel<<<(N + 1023) / 1024, 256, 0, stream>>>(hB, batch, sums, cnt, N);

    // 6. MLP head + log_softmax
    head_kernel<<<1, 128, 0, stream>>>(sums, cnt, lin1_w, lin1_b, lin2_w, lin2_b, out);
}